// MultiHeadAttention_27711128994021
// MI455X (gfx1250) — compile-verified
//
#include <hip/hip_runtime.h>
#include <hip/hip_bf16.h>

// ---------------------------------------------------------------------------
// Linear-attention MHA (elu+1 feature map) for MI455X / gfx1250.
// Heavy math = 4 fp32 GEMMs (16384x1024x1024) done with V_WMMA_F32_16X16X4_F32.
// Register-blocked 32(M) x 64(N) wave tile: 8 wmmas per 6 fragment loads
// (~10.7 FLOP/byte from cache vs 4 for a 16x16 tile). Workgroup = 8 waves
// covering 64(M) x 256(N). Everything else is memory-bound elementwise work.
// ---------------------------------------------------------------------------

#define Bc 4
#define Sc 4096
#define Hc 16
#define Dc 64
#define DMc (Hc * Dc)          // 1024
#define Mc (Bc * Sc)           // 16384
#define EPSc 1e-6f

typedef float v2f __attribute__((ext_vector_type(2)));
typedef float v8f __attribute__((ext_vector_type(8)));

// ---------------------------------------------------------------------------
// GEMM: C[m,n] = sum_k A[m,k] * W[n,k] + bias[n]   (nn.Linear: y = x @ W.T + b)
// A-frag (16x4 f32): lane L<16 holds (M=L, K=k0,k0+1), L>=16 holds (M=L-16, K=k0+2,k0+3)
// B-frag (4x16 f32): lane L<16 holds (N=L, K=k0,k0+1), L>=16 holds (N=L-16, K=k0+2,k0+3)
// C/D (16x16 f32):   vgpr r: lanes 0-15 -> (M=r, N=lane); lanes 16-31 -> (M=r+8, N=lane-16)
// ---------------------------------------------------------------------------
__global__ __launch_bounds__(256) void gemm_wmma(
    const float* __restrict__ A, const float* __restrict__ W,
    const float* __restrict__ bias, float* __restrict__ C,
    int M, int N, int K)
{
    const int wave  = threadIdx.x >> 5;   // 0..7
    const int lane  = threadIdx.x & 31;
    const int waveM = wave >> 2;          // 0..1
    const int waveN = wave & 3;           // 0..3
    const int tileM = blockIdx.y * 64 + waveM * 32;    // wave covers [tileM, tileM+32)
    const int tileN = blockIdx.x * 256 + waveN * 64;   // wave covers [tileN, tileN+64)
    const int l16   = lane & 15;
    const int hi    = lane >> 4;          // 0: K pair {0,1}; 1: K pair {2,3}

    // Per-lane row pointers (contiguous in k)
    const float* __restrict__ Ar0 = A + (size_t)(tileM + l16) * K;
    const float* __restrict__ Ar1 = Ar0 + (size_t)16 * K;
    const float* __restrict__ Wr0 = W + (size_t)(tileN + l16) * K;
    const float* __restrict__ Wr1 = Wr0 + (size_t)16 * K;
    const float* __restrict__ Wr2 = Wr0 + (size_t)32 * K;
    const float* __restrict__ Wr3 = Wr0 + (size_t)48 * K;

    v8f acc[2][4];
#pragma unroll
    for (int i = 0; i < 2; ++i)
#pragma unroll
        for (int j = 0; j < 4; ++j)
            acc[i][j] = (v8f){0.f, 0.f, 0.f, 0.f, 0.f, 0.f, 0.f, 0.f};

    for (int k0 = 0; k0 < K; k0 += 4) {
        const int kk = k0 + hi * 2;

        if ((k0 & 60) == 0) {
            // Prefetch one 64-float (256B) stride ahead on each streamed row.
            __builtin_prefetch(Ar0 + kk + 64, 0, 1);
            __builtin_prefetch(Ar1 + kk + 64, 0, 1);
            __builtin_prefetch(Wr0 + kk + 64, 0, 1);
            __builtin_prefetch(Wr1 + kk + 64, 0, 1);
            __builtin_prefetch(Wr2 + kk + 64, 0, 1);
            __builtin_prefetch(Wr3 + kk + 64, 0, 1);
        }

        v2f a0; a0.x = Ar0[kk]; a0.y = Ar0[kk + 1];
        v2f a1; a1.x = Ar1[kk]; a1.y = Ar1[kk + 1];
        v2f b0; b0.x = Wr0[kk]; b0.y = Wr0[kk + 1];
        v2f b1; b1.x = Wr1[kk]; b1.y = Wr1[kk + 1];
        v2f b2; b2.x = Wr2[kk]; b2.y = Wr2[kk + 1];
        v2f b3; b3.x = Wr3[kk]; b3.y = Wr3[kk + 1];

#if defined(__HIP_DEVICE_COMPILE__)
        acc[0][0] = __builtin_amdgcn_wmma_f32_16x16x4_f32(false, a0, false, b0, (short)0, acc[0][0], false, false);
        acc[0][1] = __builtin_amdgcn_wmma_f32_16x16x4_f32(false, a0, false, b1, (short)0, acc[0][1], false, false);
        acc[0][2] = __builtin_amdgcn_wmma_f32_16x16x4_f32(false, a0, false, b2, (short)0, acc[0][2], false, false);
        acc[0][3] = __builtin_amdgcn_wmma_f32_16x16x4_f32(false, a0, false, b3, (short)0, acc[0][3], false, false);
        acc[1][0] = __builtin_amdgcn_wmma_f32_16x16x4_f32(false, a1, false, b0, (short)0, acc[1][0], false, false);
        acc[1][1] = __builtin_amdgcn_wmma_f32_16x16x4_f32(false, a1, false, b1, (short)0, acc[1][1], false, false);
        acc[1][2] = __builtin_amdgcn_wmma_f32_16x16x4_f32(false, a1, false, b2, (short)0, acc[1][2], false, false);
        acc[1][3] = __builtin_amdgcn_wmma_f32_16x16x4_f32(false, a1, false, b3, (short)0, acc[1][3], false, false);
#else
        // Host-pass stub only: amdgcn builtins are not declared for x86.
        acc[0][0][0] += a0.x * b0.x + a1.x * b1.x + b2.x + b3.x;
#endif
    }

#pragma unroll
    for (int j = 0; j < 4; ++j) {
        const int ncol = tileN + j * 16 + l16;
        const float bv = bias[ncol];
#pragma unroll
        for (int i = 0; i < 2; ++i) {
            const int mrow = tileM + i * 16 + hi * 8;
#pragma unroll
            for (int r = 0; r < 8; ++r)
                C[(size_t)(mrow + r) * N + ncol] = acc[i][j][r] + bv;
        }
    }
}

// ---------------------------------------------------------------------------
// RoPE + phi=elu+1 + key-padding mask, computed in place over Q and K buffers.
// Also emits per-row sums: qsum[row] = sum_d phi_q, rowsum[row] = sum_d phi_k.
// One 64-thread group per (b,s,h) row; 4 rows per 256-thread block.
// ---------------------------------------------------------------------------
__global__ __launch_bounds__(256) void rope_phi(
    float* __restrict__ Q, float* __restrict__ Kb,
    const float* __restrict__ rope, const int* __restrict__ pmask,
    float* __restrict__ qsum, float* __restrict__ rowsum)
{
    __shared__ float red[512];
    const int tid = threadIdx.x;
    const int grp = tid >> 6;          // 0..3
    const int d   = tid & 63;
    const int row = blockIdx.x * 4 + grp;        // (b*S + s)*H + h
    const int bs  = row / Hc;                    // b*S + s
    const int s   = bs % Sc;
    const int b   = bs / Sc;

    const size_t idx = (size_t)row * Dc + d;
    float q = Q[idx];
    float k = Kb[idx];

    // sin_pos[d] = rope[s,0,d/2]; cos_pos[d] = rope[s,0,32 + d/2]
    const int half = d >> 1;
    const float sv = rope[(size_t)s * Dc + half];
    const float cv = rope[(size_t)s * Dc + 32 + half];

    // rot_half: even d -> -x[d+1], odd d -> +x[d-1]; pair lanes are adjacent (wave32-safe)
    const float qp  = __shfl_xor(q, 1);
    const float kp  = __shfl_xor(k, 1);
    const float sgn = (d & 1) ? 1.0f : -1.0f;
    const float rq  = q * cv + sgn * qp * sv;
    const float rk  = k * cv + sgn * kp * sv;

    // phi(x) = elu(x)+1 = x+1 (x>0) else exp(x)
    float phq = (rq > 0.f) ? (rq + 1.f) : __expf(rq);
    float phk = (rk > 0.f) ? (rk + 1.f) : __expf(rk);
    if (pmask[(size_t)b * Sc + s] == 1) phk = 0.f;

    Q[idx]  = phq;
    Kb[idx] = phk;

    red[tid]       = phq;
    red[256 + tid] = phk;
    __syncthreads();
#pragma unroll
    for (int off = 32; off > 0; off >>= 1) {
        if (d < off) {
            red[tid]       += red[tid + off];
            red[256 + tid] += red[256 + tid + off];
        }
        __syncthreads();
    }
    if (d == 0) {
        qsum[row]   = red[grp << 6];
        rowsum[row] = red[256 + (grp << 6)];
    }
}

// ---------------------------------------------------------------------------
// Per (b,h) deterministic reduction over S:
//   ksum[b,h,d]  = sum_s phi_k[b,s,h,d]
//   kvsum[b,h,v] = sum_s V[b,s,h,v] * rowsum[b,s,h]   (== sum_k kv[b,h,v,k])
// One block per (b,h); 4 sequential S-chunks combined in fixed order.
// ---------------------------------------------------------------------------
__global__ __launch_bounds__(256) void reduce_seq(
    const float* __restrict__ phiK, const float* __restrict__ V,
    const float* __restrict__ rowsum,
    float* __restrict__ ksum, float* __restrict__ kvsum)
{
    __shared__ float red[512];
    const int bh = blockIdx.x;            // b*H + h
    const int b  = bh / Hc;
    const int h  = bh % Hc;
    const int tid = threadIdx.x;
    const int d   = tid & 63;
    const int c   = tid >> 6;             // chunk 0..3

    float pk = 0.f, pkv = 0.f;
    const int s0 = c * (Sc / 4);
    const int s1 = s0 + (Sc / 4);
    for (int s = s0; s < s1; ++s) {
        const size_t row = (size_t)(b * Sc + s) * Hc + h;
        const size_t idx = row * Dc + d;
        pk  += phiK[idx];
        pkv += V[idx] * rowsum[row];
    }
    red[tid]       = pk;
    red[256 + tid] = pkv;
    __syncthreads();
    if (c == 0) {
        const float tk = red[d] + red[64 + d] + red[128 + d] + red[192 + d];
        const float tv = red[256 + d] + red[320 + d] + red[384 + d] + red[448 + d];
        ksum[(size_t)bh * Dc + d]  = tk;
        kvsum[(size_t)bh * Dc + d] = tv;
    }
}

// ---------------------------------------------------------------------------
// z = 1/max(phi_q . ksum, EPS);  A[b,s,h*D+v] = qsum * z * kvsum[b,h,v]
// ---------------------------------------------------------------------------
__global__ __launch_bounds__(256) void z_attn(
    const float* __restrict__ phiQ, const float* __restrict__ qsum,
    const float* __restrict__ ksum, const float* __restrict__ kvsum,
    float* __restrict__ Amat)
{
    __shared__ float red[256];
    const int tid = threadIdx.x;
    const int grp = tid >> 6;
    const int d   = tid & 63;
    const int row = blockIdx.x * 4 + grp;        // (b*S + s)*H + h
    const int h   = row % Hc;
    const int b   = (row / Hc) / Sc;
    const int bh  = b * Hc + h;

    const size_t idx = (size_t)row * Dc + d;
    red[tid] = phiQ[idx] * ksum[(size_t)bh * Dc + d];
    __syncthreads();
#pragma unroll
    for (int off = 32; off > 0; off >>= 1) {
        if (d < off) red[tid] += red[tid + off];
        __syncthreads();
    }
    const float dot = red[grp << 6];
    const float z   = 1.f / fmaxf(dot, EPSc);
    Amat[idx] = qsum[row] * z * kvsum[(size_t)bh * Dc + d];
}

// ---------------------------------------------------------------------------
// Launch pipeline. Workspace layout (bytes):
//   [0, MAT)        Q  -> becomes phi_q
//   [MAT, 2MAT)     K  -> becomes phi_k -> becomes attn matrix A
//   [2MAT, 3MAT)    V
//   then qsum (B*S*H), rowsum (B*S*H), ksum (B*H*D), kvsum (B*H*D)
// ---------------------------------------------------------------------------
extern "C" void kernel_launch(void* const* d_in, const int* in_sizes, int n_in,
                              void* d_out, int out_size, void* d_ws, size_t ws_size,
                              hipStream_t stream) {
    const float* x    = (const float*)d_in[0];
    const float* rope = (const float*)d_in[1];
    const int*   pm   = (const int*)d_in[2];
    const float* Wq   = (const float*)d_in[3];
    const float* bq   = (const float*)d_in[4];
    const float* Wk   = (const float*)d_in[5];
    const float* bk   = (const float*)d_in[6];
    const float* Wv   = (const float*)d_in[7];
    const float* bv   = (const float*)d_in[8];
    const float* Wo   = (const float*)d_in[9];
    const float* bo   = (const float*)d_in[10];
    float* out = (float*)d_out;

    char* ws = (char*)d_ws;
    const size_t MAT = (size_t)Mc * DMc * sizeof(float);       // 64 MB
    const size_t R   = (size_t)Bc * Sc * Hc;                   // 262144 rows
    float* Q      = (float*)(ws);
    float* K      = (float*)(ws + MAT);
    float* V      = (float*)(ws + 2 * MAT);
    float* qsum   = (float*)(ws + 3 * MAT);
    float* rowsum = (float*)(ws + 3 * MAT + R * sizeof(float));
    float* ksum   = (float*)(ws + 3 * MAT + 2 * R * sizeof(float));
    float* kvsum  = (float*)(ws + 3 * MAT + 2 * R * sizeof(float) + (size_t)Bc * Hc * Dc * sizeof(float));

    const dim3 ggrid(DMc / 256, Mc / 64);   // (4, 256)
    const dim3 gblk(256);

    // Q/K/V projections (fp32 WMMA)
    gemm_wmma<<<ggrid, gblk, 0, stream>>>(x, Wq, bq, Q, Mc, DMc, DMc);
    gemm_wmma<<<ggrid, gblk, 0, stream>>>(x, Wk, bk, K, Mc, DMc, DMc);
    gemm_wmma<<<ggrid, gblk, 0, stream>>>(x, Wv, bv, V, Mc, DMc, DMc);

    // RoPE + phi (+ row sums), in place over Q/K
    rope_phi<<<(int)(R / 4), 256, 0, stream>>>(Q, K, rope, pm, qsum, rowsum);

    // Deterministic reductions over S
    reduce_seq<<<Bc * Hc, 256, 0, stream>>>(K, V, rowsum, ksum, kvsum);

    // z + attention matrix (written over K buffer)
    z_attn<<<(int)(R / 4), 256, 0, stream>>>(Q, qsum, ksum, kvsum, K);

    // Output projection
    gemm_wmma<<<ggrid, gblk, 0, stream>>>(K, Wo, bo, out, Mc, DMc, DMc);
}